// SpatialAttention_13314398617934
// MI455X (gfx1250) — compile-verified
//
#include <hip/hip_runtime.h>
#include <math.h>

#define B_    32
#define C_    64
#define T_    8192
#define OUT_  129
#define OPAD_ 144          // ws row stride (stages 1-2 layout)
#define KF_   32
#define F_    (KF_ * KF_)  // 1024 frequencies
#define TWO_PI_ 6.283185307179586f
#define DROP_ 0.1f

typedef __attribute__((ext_vector_type(2))) float v2f;
typedef __attribute__((ext_vector_type(8))) float v8f;

// ---------------------------------------------------------------------------
// Stage 1: logits[b][o][c] = sum_f zr[o][f]*cos(phase[b,c,f]) + zi[o][f]*sin(...)
// One block per (c, b). cos/sin table built in LDS, then 129 dot products.
// ---------------------------------------------------------------------------
__global__ void __launch_bounds__(128)
spatial_logits(const float* __restrict__ pos,
               const float* __restrict__ zr,
               const float* __restrict__ zi,
               float* __restrict__ logits) {
  __shared__ float cs[F_];
  __shared__ float sn[F_];
  const int c = blockIdx.x;
  const int b = blockIdx.y;
  const float x = pos[(b * C_ + c) * 2 + 0];
  const float y = pos[(b * C_ + c) * 2 + 1];

  for (int f = threadIdx.x; f < F_; f += 128) {
    const float kk = (float)((f & (KF_ - 1)) + 1);   // varies fastest
    const float ll = (float)((f >> 5) + 1);
    float s, co;
    __sincosf(TWO_PI_ * (x * kk + y * ll), &s, &co);
    cs[f] = co;
    sn[f] = s;
  }
  __syncthreads();

  for (int o = threadIdx.x; o < OUT_; o += 128) {
    const float* __restrict__ zrr = zr + o * F_;
    const float* __restrict__ zii = zi + o * F_;
    float acc = 0.0f;
    #pragma unroll 8
    for (int f = 0; f < F_; ++f)
      acc = fmaf(zrr[f], cs[f], fmaf(zii[f], sn[f], acc));
    logits[(b * OPAD_ + o) * C_ + c] = acc;
  }
}

// ---------------------------------------------------------------------------
// Stage 2: channel-drop mask + softmax over c (64). One 64-thread block per (o, b).
// ---------------------------------------------------------------------------
__global__ void __launch_bounds__(64)
spatial_softmax(const float* __restrict__ pos,
                const int* __restrict__ drop,
                float* __restrict__ attn) {
  __shared__ float red[C_];
  const int o = blockIdx.x;
  const int b = blockIdx.y;
  const int c = threadIdx.x;

  const int di    = drop[b];
  const float pdx = pos[(b * C_ + di) * 2 + 0];
  const float pdy = pos[(b * C_ + di) * 2 + 1];
  const float dx  = pos[(b * C_ + c) * 2 + 0] - pdx;
  const float dy  = pos[(b * C_ + c) * 2 + 1] - pdy;
  const bool keep = (dx * dx + dy * dy) >= DROP_;

  float v = attn[(b * OPAD_ + o) * C_ + c];
  v = keep ? v : -1e9f;

  red[c] = v;
  __syncthreads();
  for (int s = 32; s > 0; s >>= 1) {
    if (c < s) red[c] = fmaxf(red[c], red[c + s]);
    __syncthreads();
  }
  const float m = red[0];
  __syncthreads();

  const float e = __expf(v - m);
  red[c] = e;
  __syncthreads();
  for (int s = 32; s > 0; s >>= 1) {
    if (c < s) red[c] += red[c + s];
    __syncthreads();
  }
  attn[(b * OPAD_ + o) * C_ + c] = e / red[0];
}

// ---------------------------------------------------------------------------
// Stage 3: out[b][o][t] = sum_c attn[b][o][c] * X[b][c][t] via
// V_WMMA_F32_16X16X4_F32. Block: 8 waves; wave w owns t-strip [t0+16w, +16).
// Each wave preloads all 16 B-fragments of X (full K=64) into registers once
// (non-temporal: read-once traffic), then runs 8 full 16-row o-tiles with
// unconditional NT stores addressed as base + r*T_ so the 32KB row stride
// lands in the 24-bit store immediate. Leftover row o=128 is a length-64 dot
// on the resident bf[] fragments plus one cross-half shuffle.
// ---------------------------------------------------------------------------
__global__ void __launch_bounds__(256)
spatial_attn_gemm(const float* __restrict__ Xg,
                  const float* __restrict__ attn,
                  float* __restrict__ out) {
  __shared__ float a_lds[OUT_ * C_];   // 129*64 floats = 33 KB

  const int b    = blockIdx.y;
  const int t0   = blockIdx.x * 128;
  const int tid  = threadIdx.x;
  const int w    = tid >> 5;
  const int lane = tid & 31;

  // Stage the 129 real attn rows for this batch.
  for (int idx = tid; idx < OUT_ * C_; idx += 256)
    a_lds[idx] = attn[b * OPAD_ * C_ + idx];
  __syncthreads();

  // B-fragment layout (4x16 fp32, 2 VGPRs): elem0 -> K=c0 (lanes 0-15) / K=c0+2
  // (lanes 16-31); elem1 -> K=c0+1 / K=c0+3. N = lane & 15.
  const int tcol = t0 + w * 16 + (lane & 15);
  const int rsel = (lane < 16) ? 0 : 2;

  // All 32 X loads off one per-lane base with constant row-stride offsets.
  const float* __restrict__ xb = Xg + (size_t)(b * C_ + rsel) * T_ + tcol;
  v2f bf[16];
  #pragma unroll
  for (int kk = 0; kk < 16; ++kk) {
    bf[kk].x = __builtin_nontemporal_load(xb + (size_t)(kk * 4) * T_);
    bf[kk].y = __builtin_nontemporal_load(xb + (size_t)(kk * 4 + 1) * T_);
  }

  const int m   = lane & 15;
  const int mhi = (lane < 16) ? 0 : 8;

  // Uniform scalar base; per-tile one 32-bit VGPR offset; per-row constant
  // immediate r*T_ (r*32KB, fits the signed 24-bit IOFFSET).
  float* __restrict__ outb = out + (size_t)b * OUT_ * T_;

  // 8 full o-tiles: rows 0..127, always in range -> unconditional stores.
  #pragma unroll 1
  for (int ot = 0; ot < 8; ++ot) {
    v8f acc = {};
    // A-fragment (16x4 fp32, 2 VGPRs): elem0 -> K=0 (lanes 0-15) / K=2
    // (lanes 16-31); elem1 -> K=1 / K=3. M = lane & 15.
    #pragma unroll
    for (int kk = 0; kk < 16; ++kk) {
      const int kb = kk * 4 + rsel;
      v2f a;
      a.x = a_lds[(ot * 16 + m) * C_ + kb];
      a.y = a_lds[(ot * 16 + m) * C_ + kb + 1];
      acc = __builtin_amdgcn_wmma_f32_16x16x4_f32(
          /*neg_a=*/false, a, /*neg_b=*/false, bf[kk],
          /*c_mod=*/(short)0, acc, /*reuse_a=*/false, /*reuse_b=*/false);
    }
    // D layout: VGPR r -> M = r (lanes 0-15) / r+8 (lanes 16-31), N = lane&15.
    const unsigned voff = (unsigned)((ot * 16 + mhi) * T_ + tcol);
    float* __restrict__ p = outb + voff;
    #pragma unroll
    for (int r = 0; r < 8; ++r)
      __builtin_nontemporal_store(acc[r], p + r * T_);
  }

  // Tail row o = 128: lanes n and n^16 share tcol and together cover all 64
  // channels in their bf[] fragments. Partial dot per lane, combine across
  // halves with one shuffle, lanes 0-15 store.
  {
    const float* __restrict__ arow = &a_lds[128 * C_];
    float partial = 0.0f;
    #pragma unroll
    for (int kk = 0; kk < 16; ++kk) {
      const int kb = kk * 4 + rsel;
      partial = fmaf(arow[kb],     bf[kk].x, partial);
      partial = fmaf(arow[kb + 1], bf[kk].y, partial);
    }
    const float full = partial + __shfl_xor(partial, 16, 32);
    if (lane < 16)
      __builtin_nontemporal_store(full, outb + (unsigned)(128 * T_ + tcol));
  }
}

// ---------------------------------------------------------------------------
extern "C" void kernel_launch(void* const* d_in, const int* in_sizes, int n_in,
                              void* d_out, int out_size, void* d_ws, size_t ws_size,
                              hipStream_t stream) {
  const float* X   = (const float*)d_in[0];  // [B, C, T]
  const float* pos = (const float*)d_in[1];  // [B, C, 2]
  const int*   di  = (const int*)d_in[2];    // [B]
  const float* zr  = (const float*)d_in[3];  // [OUT, K*K]
  const float* zi  = (const float*)d_in[4];  // [OUT, K*K]
  float* out  = (float*)d_out;               // [B, OUT, T]
  float* attn = (float*)d_ws;                // [B, OPAD, C] scratch (~1.13 MB)

  spatial_logits<<<dim3(C_, B_), 128, 0, stream>>>(pos, zr, zi, attn);
  spatial_softmax<<<dim3(OUT_, B_), C_, 0, stream>>>(pos, di, attn);
  spatial_attn_gemm<<<dim3(T_ / 128, B_), 256, 0, stream>>>(X, attn, out);
}